// RuleModel_81922206204617
// MI455X (gfx1250) — compile-verified
//
#include <hip/hip_runtime.h>
#include <stdint.h>
#include <math.h>

#define TPB 256
#define ACH 32  // argmax din-chunks for parallelism

// Padded conceptual row layout in LDS (16B-aligned segments), width 5124:
//   [0,512)=x  [512,1024)=1-x  [1024]=1 [1025]=0  [1026,1028)=pad
//   [1028,2052)=c1  [2052,3076)=d1  [3076,4100)=c2  [4100,5124)=d2
// Conceptual index k (reference layout) maps to k + (k>=1026 ? 2 : 0).

// ---------------------------------------------------------------------------
// CDNA5 async global->LDS copy (ASYNCcnt-tracked).
// ---------------------------------------------------------------------------
__device__ __forceinline__ void async_copy_b128(void* lds_dst, const void* gsrc) {
  uint32_t l = (uint32_t)(uintptr_t)lds_dst;
  asm volatile("global_load_async_to_lds_b128 %0, %1, off"
               :: "v"(l), "v"(gsrc)
               : "memory");
}

__device__ __forceinline__ void wait_async_all() {
  asm volatile("s_wait_asynccnt 0x0" ::: "memory");
}

// ---------------------------------------------------------------------------
// Phase 1: per-chunk argmax + online sum-exp over rows (axis 0) of (w+g)/tau.
// ---------------------------------------------------------------------------
__global__ void gate_argmax_part(const float* __restrict__ w,
                                 const float* __restrict__ g,
                                 int din, int dout, const int* __restrict__ taup,
                                 float* __restrict__ pv, int* __restrict__ pi,
                                 float* __restrict__ ps) {
  int j = blockIdx.x * blockDim.x + threadIdx.x;
  int ch = blockIdx.y;
  if (j >= dout) return;
  const float invt = 1.0f / (float)(*taup);
  int i0 = (int)(((long long)din * ch) / ACH);
  int i1 = (int)(((long long)din * (ch + 1)) / ACH);
  float m = -INFINITY, s = 0.0f;
  int bi = i0;
  for (int i = i0; i < i1; ++i) {
    float v = w[(size_t)i * dout + j] + g[(size_t)i * dout + j];
    if (v > m) {                            // strict > => first occurrence wins
      s = s * expf((m - v) * invt) + 1.0f;  // s==0 first time: 0*0+1
      m = v;
      bi = i;
    } else {
      s += expf((v - m) * invt);
    }
  }
  size_t o = (size_t)ch * dout + j;
  pv[o] = m;
  pi[o] = bi;
  ps[o] = s;
}

// Phase 2: combine chunks; emit argmax index + ST gate scale fl((1+y*)-y*).
__global__ void gate_argmax_final(const float* __restrict__ pv,
                                  const int* __restrict__ pi,
                                  const float* __restrict__ ps,
                                  int dout, const int* __restrict__ taup,
                                  int* __restrict__ idx, float* __restrict__ scl) {
  int j = blockIdx.x * blockDim.x + threadIdx.x;
  if (j >= dout) return;
  const float invt = 1.0f / (float)(*taup);
  float gmax = -INFINITY;
  int bi = 0;
  for (int c = 0; c < ACH; ++c) {
    float v = pv[(size_t)c * dout + j];
    if (v > gmax) { gmax = v; bi = pi[(size_t)c * dout + j]; }
  }
  float total = 0.0f;
  for (int c = 0; c < ACH; ++c)
    total += ps[(size_t)c * dout + j] * expf((pv[(size_t)c * dout + j] - gmax) * invt);
  float ystar = 1.0f / total;           // softmax prob at the argmax
  idx[j] = bi;
  scl[j] = (1.0f + ystar) - ystar;      // forward value of hard + y - stop_grad(y)
}

// ---------------------------------------------------------------------------
// Fully fused network: all 3 logic layers + final projection, entirely in LDS.
// Each block owns ROWS rows; no intermediate ever touches global memory.
// ---------------------------------------------------------------------------
template <int ROWS>
__global__ __launch_bounds__(TPB) void fused_logic_kernel(
    const float* __restrict__ x,
    const int* __restrict__ idx1, const float* __restrict__ scl1,
    const int* __restrict__ idx2, const float* __restrict__ scl2,
    const int* __restrict__ idx3, const float* __restrict__ scl3,
    const float* __restrict__ lin_w,
    float* __restrict__ out) {
  constexpr int DP = 5124;
  __shared__ __align__(16) float tile[ROWS][DP];
  __shared__ __align__(16) float lw[1024];
  __shared__ float red[ROWS * TPB];

  const int tid = threadIdx.x;
  const int row0 = blockIdx.x * ROWS;

  // ---- async stage: x rows (512 floats = 128 b128 each) + lin_w ----
  for (int r = 0; r < ROWS; ++r) {
    if (tid < 128)
      async_copy_b128(&tile[r][4 * tid], x + (size_t)(row0 + r) * 512 + 4 * tid);
  }
  async_copy_b128(&lw[4 * tid], lin_w + 4 * tid);  // 1024 floats = 256 b128
  wait_async_all();
  __syncthreads();

  // ---- input augmentation: [512,1024)=1-x ; [1024]=1 ; [1025]=0 ----
  for (int r = 0; r < ROWS; ++r) {
    tile[r][512 + tid] = 1.0f - tile[r][tid];
    tile[r][768 + tid] = 1.0f - tile[r][256 + tid];
  }
  if (tid < ROWS) {
    tile[tid][1024] = 1.0f;
    tile[tid][1025] = 0.0f;
  }
  __syncthreads();

  // ---- layer 1: reads [0,1026) (identity map), writes c1/d1 [1028,3076) ----
  {
    int i1a[4], i1b[4], i1c[4], i1d[4];
    float s1a[4], s1b[4], s1c[4], s1d[4];
#pragma unroll
    for (int u = 0; u < 4; ++u) {
      int j = tid + u * TPB;
      i1a[u] = idx1[j];        i1b[u] = idx1[1024 + j];
      i1c[u] = idx1[2048 + j]; i1d[u] = idx1[3072 + j];
      s1a[u] = scl1[j];        s1b[u] = scl1[1024 + j];
      s1c[u] = scl1[2048 + j]; s1d[u] = scl1[3072 + j];
    }
#pragma unroll
    for (int r = 0; r < ROWS; ++r) {
#pragma unroll
      for (int u = 0; u < 4; ++u) {
        int j = tid + u * TPB;
        float c = (tile[r][i1a[u]] * s1a[u]) * (tile[r][i1b[u]] * s1b[u]);
        float a = tile[r][i1c[u]] * s1c[u];
        float b = tile[r][i1d[u]] * s1d[u];
        float d = 1.0f - (1.0f - a) * (1.0f - b);
        tile[r][1028 + j] = c;   // write segment disjoint from read segment
        tile[r][2052 + j] = d;
      }
    }
  }
  __syncthreads();

  // ---- layer 2: reads [0,3076), writes c2/d2 [3076,5124) ----
  {
    int i2a[4], i2b[4], i2c[4], i2d[4];
    float s2a[4], s2b[4], s2c[4], s2d[4];
#pragma unroll
    for (int u = 0; u < 4; ++u) {
      int j = tid + u * TPB;
      int k;
      k = idx2[j];        i2a[u] = (k >= 1026) ? k + 2 : k;
      k = idx2[1024 + j]; i2b[u] = (k >= 1026) ? k + 2 : k;
      k = idx2[2048 + j]; i2c[u] = (k >= 1026) ? k + 2 : k;
      k = idx2[3072 + j]; i2d[u] = (k >= 1026) ? k + 2 : k;
      s2a[u] = scl2[j];        s2b[u] = scl2[1024 + j];
      s2c[u] = scl2[2048 + j]; s2d[u] = scl2[3072 + j];
    }
#pragma unroll
    for (int r = 0; r < ROWS; ++r) {
#pragma unroll
      for (int u = 0; u < 4; ++u) {
        int j = tid + u * TPB;
        float c = (tile[r][i2a[u]] * s2a[u]) * (tile[r][i2b[u]] * s2b[u]);
        float a = tile[r][i2c[u]] * s2c[u];
        float b = tile[r][i2d[u]] * s2d[u];
        float d = 1.0f - (1.0f - a) * (1.0f - b);
        tile[r][3076 + j] = c;
        tile[r][4100 + j] = d;
      }
    }
  }
  __syncthreads();

  // ---- layer 3 (DOUT=512) + fused lin_w projection + reduction ----
  {
    int i3a[2], i3b[2], i3c[2], i3d[2];
    float s3a[2], s3b[2], s3c[2], s3d[2];
#pragma unroll
    for (int u = 0; u < 2; ++u) {
      int j = tid + u * TPB;
      int k;
      k = idx3[j];        i3a[u] = (k >= 1026) ? k + 2 : k;
      k = idx3[512 + j];  i3b[u] = (k >= 1026) ? k + 2 : k;
      k = idx3[1024 + j]; i3c[u] = (k >= 1026) ? k + 2 : k;
      k = idx3[1536 + j]; i3d[u] = (k >= 1026) ? k + 2 : k;
      s3a[u] = scl3[j];        s3b[u] = scl3[512 + j];
      s3c[u] = scl3[1024 + j]; s3d[u] = scl3[1536 + j];
    }
#pragma unroll
    for (int r = 0; r < ROWS; ++r) {
      float racc = 0.0f;
#pragma unroll
      for (int u = 0; u < 2; ++u) {
        int j = tid + u * TPB;
        float c = (tile[r][i3a[u]] * s3a[u]) * (tile[r][i3b[u]] * s3b[u]);
        float a = tile[r][i3c[u]] * s3c[u];
        float b = tile[r][i3d[u]] * s3d[u];
        float d = 1.0f - (1.0f - a) * (1.0f - b);
        racc += lw[j] * c + lw[512 + j] * d;   // h3 = [c | d], lin_w fused
      }
      red[r * TPB + tid] = racc;
    }
  }
  __syncthreads();

  // deterministic tree reduction per row
#pragma unroll
  for (int s = TPB / 2; s > 0; s >>= 1) {
    if (tid < s) {
#pragma unroll
      for (int r = 0; r < ROWS; ++r)
        red[r * TPB + tid] += red[r * TPB + tid + s];
    }
    __syncthreads();
  }
  if (tid < ROWS) out[row0 + tid] = red[tid * TPB];
}

// ---------------------------------------------------------------------------
// Inputs (setup_inputs dict order):
//  [0..7]  l1: wc1,wc2,wd1,wd2,gc1,gc2,gd1,gd2   (1026x1024 each)
//  [8..15] l2: same                               (3074x1024)
//  [16..23] l3: same                              (5122x512)
//  [24] x (32768x512)  [25] lin_w (1x1024)  [26] tau (int scalar)
// ---------------------------------------------------------------------------
extern "C" void kernel_launch(void* const* d_in, const int* in_sizes, int n_in,
                              void* d_out, int out_size, void* d_ws, size_t ws_size,
                              hipStream_t stream) {
  (void)in_sizes; (void)n_in; (void)out_size; (void)ws_size;
  const float* x = (const float*)d_in[24];
  const float* lin_w = (const float*)d_in[25];
  const int* taup = (const int*)d_in[26];

  // Workspace (<1MB total now; no HBM intermediates):
  //  [0): idx1(4096 i32) [16K): idx2(4096) [32K): idx3(2048)
  //  [40K): scl1(4096 f32) [56K): scl2(4096) [72K): scl3(2048)
  //  [256K): pv (128K)  [384K): pi (128K)  [512K): ps (128K)
  char* ws = (char*)d_ws;
  int* idx1 = (int*)(ws + (0 << 10));
  int* idx2 = (int*)(ws + (16 << 10));
  int* idx3 = (int*)(ws + (32 << 10));
  float* scl1 = (float*)(ws + (40 << 10));
  float* scl2 = (float*)(ws + (56 << 10));
  float* scl3 = (float*)(ws + (72 << 10));
  float* pv = (float*)(ws + (256 << 10));
  int* pi = (int*)(ws + (384 << 10));
  float* ps = (float*)(ws + (512 << 10));

  const int din[3] = {1026, 3074, 5122};
  const int dout[3] = {1024, 1024, 512};
  int* idxL[3] = {idx1, idx2, idx3};
  float* sclL[3] = {scl1, scl2, scl3};

  for (int li = 0; li < 3; ++li) {
    int cb = (dout[li] + TPB - 1) / TPB;
    for (int gi = 0; gi < 4; ++gi) {
      const float* w = (const float*)d_in[li * 8 + gi];
      const float* g = (const float*)d_in[li * 8 + 4 + gi];
      gate_argmax_part<<<dim3(cb, ACH), TPB, 0, stream>>>(
          w, g, din[li], dout[li], taup, pv, pi, ps);
      gate_argmax_final<<<cb, TPB, 0, stream>>>(
          pv, pi, ps, dout[li], taup, idxL[li] + gi * dout[li], sclL[li] + gi * dout[li]);
    }
  }

  fused_logic_kernel<2><<<32768 / 2, TPB, 0, stream>>>(
      x, idx1, scl1, idx2, scl2, idx3, scl3, lin_w, (float*)d_out);
}